// SpanGCNModule_74594991997486
// MI455X (gfx1250) — compile-verified
//
#include <hip/hip_runtime.h>

// Problem constants (from reference)
#define B_   32
#define S_   2048
#define H_   768
#define M_   64
#define NC_  128

typedef __attribute__((ext_vector_type(2))) float v2f;
typedef __attribute__((ext_vector_type(8))) float v8f;

__device__ __forceinline__ int imax(int a, int b) { return a > b ? a : b; }

// -----------------------------------------------------------------------------
// Kernel A: pred_repr[b, :] = mean of x[b, ps:pe, :].  One block per batch.
// 192 threads; each owns one float4 of H=768. ~3 MB total reads.
// -----------------------------------------------------------------------------
__global__ __launch_bounds__(192) void pred_mean_kernel(
    const float* __restrict__ x,
    const int*   __restrict__ pred_start,
    const int*   __restrict__ pred_end,
    float*       __restrict__ pred_repr)       // [B_, H_]
{
    const int b  = blockIdx.x;
    const int h4 = threadIdx.x << 2;
    const float4* xb = reinterpret_cast<const float4*>(x + (size_t)b * S_ * H_);

    const int p0 = pred_start[b];
    const int p1 = pred_end[b];
    float4 acc = make_float4(0.f, 0.f, 0.f, 0.f);
    for (int r = p0; r < p1; ++r) {
        float4 v = xb[(r * H_ + h4) >> 2];
        acc.x += v.x; acc.y += v.y; acc.z += v.z; acc.w += v.w;
    }
    const float inv = 1.0f / (float)imax(1, p1 - p0);
    reinterpret_cast<float4*>(pred_repr + (size_t)b * H_)[h4 >> 2] =
        make_float4(acc.x * inv, acc.y * inv, acc.z * inv, acc.w * inv);
}

// -----------------------------------------------------------------------------
// Kernel B: span_repr[bm, :] = mean of x[b, s:e, :].  One block per (b,m).
// Dominant phase: ~200 MB of row reads; span rows live in [0,1056) so the
// working set (~104 MB) is L2-resident (192 MB) across overlapping spans.
// -----------------------------------------------------------------------------
__global__ __launch_bounds__(192) void span_mean_kernel(
    const float* __restrict__ x,
    const int*   __restrict__ start_idx,
    const int*   __restrict__ end_idx,
    float*       __restrict__ span_repr)       // [B_*M_, H_]
{
    const int bm = blockIdx.x;          // 0 .. B_*M_-1
    const int b  = bm >> 6;             // / M_ (M_ == 64)
    const int h4 = threadIdx.x << 2;
    const float4* xb = reinterpret_cast<const float4*>(x + (size_t)b * S_ * H_);

    const int s0 = start_idx[bm];
    const int e0 = end_idx[bm];
    float4 acc = make_float4(0.f, 0.f, 0.f, 0.f);
    for (int r = s0; r < e0; ++r) {
        float4 v = xb[(r * H_ + h4) >> 2];
        acc.x += v.x; acc.y += v.y; acc.z += v.z; acc.w += v.w;
    }
    const float inv = 1.0f / (float)imax(1, e0 - s0);
    reinterpret_cast<float4*>(span_repr + (size_t)bm * H_)[h4 >> 2] =
        make_float4(acc.x * inv, acc.y * inv, acc.z * inv, acc.w * inv);
}

// -----------------------------------------------------------------------------
// Generic fp32 WMMA GEMM: out[rows,128] = A[rows,K] x W[K,128] + addend.
// One wave32 per 16-row x 128-col strip; 8 accumulator tiles so the A fragment
// is loaded once per K-step and reused 8x. Grids are sized exactly one row
// tile per wave (no divergence; EXEC all-1s as WMMA requires).
//
// Epilogue addend is a per-row-group vector: addend[(row0 >> addend_shift)][col]
//   - pred GEMM : addend = b_label (1 row),    addend_shift = 31 -> row 0
//   - main GEMM : addend = predlogits (32 rows), addend_shift = 6 -> b = bm>>6
// (16-row tiles never cross a 64-row batch group, so the index is tile-uniform.)
//
// Operand layouts per CDNA5 ISA 7.12.2 (wave32), V_WMMA_F32_16X16X4_F32:
//   A 16x4 f32 : lane L -> row (L&15), VGPR0/1 = K = (L>>4)*2, (L>>4)*2+1
//   B 4x16 f32 : lane L -> col (L&15), VGPR0/1 = K = (L>>4)*2, (L>>4)*2+1
//   C/D 16x16  : lane L -> col (L&15), VGPR r  = row r + (L>>4)*8
// -----------------------------------------------------------------------------
__global__ __launch_bounds__(128) void gemm_wmma_kernel(
    const float* __restrict__ A,       // [rows, K]
    const float* __restrict__ W,       // [K, 128]
    const float* __restrict__ addend,  // [rows >> addend_shift (+1), 128]
    float*       __restrict__ out,     // [rows, 128]
    int K, int addend_shift)
{
    const int wavesPerBlk = blockDim.x >> 5;
    const int wave  = blockIdx.x * wavesPerBlk + (threadIdx.x >> 5);
    const int lane  = threadIdx.x & 31;
    const int row0  = wave << 4;            // first of 16 output rows
    const int arow  = lane & 15;
    const int khalf = (lane >> 4) << 1;     // 0 or 2
    const int bcol  = lane & 15;

    v8f acc[8];
    const v8f vzero = {0.f, 0.f, 0.f, 0.f, 0.f, 0.f, 0.f, 0.f};
#pragma unroll
    for (int n = 0; n < 8; ++n) acc[n] = vzero;

    const float* Arow = A + (size_t)(row0 + arow) * K;

    for (int k = 0; k < K; k += 4) {
        v2f afrag;
        afrag.x = Arow[k + khalf];
        afrag.y = Arow[k + khalf + 1];
        const float* Wk = W + (size_t)(k + khalf) * NC_;
#pragma unroll
        for (int n = 0; n < 8; ++n) {
            v2f bfrag;
            bfrag.x = Wk[(n << 4) + bcol];          // K = k+khalf
            bfrag.y = Wk[NC_ + (n << 4) + bcol];    // K = k+khalf+1
            acc[n] = __builtin_amdgcn_wmma_f32_16x16x4_f32(
                /*neg_a=*/false, afrag,
                /*neg_b=*/false, bfrag,
                /*c_mod=*/(short)0, acc[n],
                /*reuse_a=*/false, /*reuse_b=*/false);
        }
    }

    // Epilogue: fuse per-group addend (bias or predicate logits).
    const float* addrow = addend + (size_t)((unsigned)row0 >> addend_shift) * NC_;
    const int rbase = row0 + ((lane >> 4) << 3);
#pragma unroll
    for (int n = 0; n < 8; ++n) {
        const int col = (n << 4) + bcol;
        const float av = addrow[col];
#pragma unroll
        for (int r = 0; r < 8; ++r) {
            out[(size_t)(rbase + r) * NC_ + col] = acc[n][r] + av;
        }
    }
}

// -----------------------------------------------------------------------------
// Launch. Inputs (setup_inputs() order):
//   0: input [B,S,H] f32        1: start_index [B,M] i32   2: end_index [B,M] i32
//   3: predicate_start [B] i32  4: predicate_end [B] i32
//   5: W_label [2H,NC] f32      6: b_label [NC] f32
// Output: logits [B,M,NC] f32.
// Workspace: span_repr [2048,768] | pred_repr [32,768] | predlogits [32,128]
// -----------------------------------------------------------------------------
extern "C" void kernel_launch(void* const* d_in, const int* in_sizes, int n_in,
                              void* d_out, int out_size, void* d_ws, size_t ws_size,
                              hipStream_t stream)
{
    const float* x     = (const float*)d_in[0];
    const int*   sidx  = (const int*)d_in[1];
    const int*   eidx  = (const int*)d_in[2];
    const int*   ps    = (const int*)d_in[3];
    const int*   pe    = (const int*)d_in[4];
    const float* W     = (const float*)d_in[5];   // [1536, 128]
    const float* bvec  = (const float*)d_in[6];   // [128]
    float* out = (float*)d_out;

    float* span_repr  = (float*)d_ws;                              // [2048, 768]
    float* pred_repr  = span_repr + (size_t)(B_ * M_) * H_;        // [32, 768]
    float* predlogits = pred_repr + (size_t)B_ * H_;               // [32, 128]

    const float* W_top = W;                 // rows 0..767   (span part)
    const float* W_bot = W + (size_t)H_ * NC_;  // rows 768..1535 (predicate part)

    // Phase 1: pooling (independent kernels, stream-ordered).
    pred_mean_kernel<<<B_, 192, 0, stream>>>(x, ps, pe, pred_repr);
    span_mean_kernel<<<B_ * M_, 192, 0, stream>>>(x, sidx, eidx, span_repr);

    // Phase 2a: predlogits[32,128] = pred_repr x W_bot + b_label.
    // 32 rows -> 2 row tiles -> 2 waves -> one 64-thread block.
    gemm_wmma_kernel<<<1, 64, 0, stream>>>(pred_repr, W_bot, bvec, predlogits,
                                           H_, /*addend_shift=*/31);

    // Phase 2b: out[2048,128] = span_repr x W_top + predlogits[bm>>6].
    // 128 row tiles -> 128 waves -> 32 blocks x 4 waves.
    gemm_wmma_kernel<<<32, 128, 0, stream>>>(span_repr, W_top, predlogits, out,
                                             H_, /*addend_shift=*/6);
}